// NewModel_66176856097442
// MI455X (gfx1250) — compile-verified
//
#include <hip/hip_runtime.h>
#include <hip/hip_bf16.h>

// CDNA5 / gfx1250: wave32, WMMA 16x16x4 f32.
typedef __attribute__((ext_vector_type(2))) float v2f;
typedef __attribute__((ext_vector_type(8))) float v8f;

#define DIM 128
#define MARGIN 1.0f
#define BLOCK 256              // 8 waves -> 128 samples per block
#define SAMPLES_PER_BLOCK 128

// acc += A(16x4 squared-diffs) * B(4x16 ones)  -> every column of acc holds the
// running per-row sum of squares.  Full f32 precision.
__device__ __forceinline__ v8f wmma_sumsq(v2f a, v2f ones, v8f c) {
    return __builtin_amdgcn_wmma_f32_16x16x4_f32(
        /*neg_a=*/false, a, /*neg_b=*/false, ones,
        /*c_mod=*/(short)0, c, /*reuse_a=*/false, /*reuse_b=*/false);
}

__global__ __launch_bounds__(BLOCK) void crt_cost_kernel(
    const int* __restrict__ lIdx,  const int* __restrict__ rIdx,
    const int* __restrict__ relIdx,
    const int* __restrict__ nlIdx, const int* __restrict__ nrIdx,
    const float* __restrict__ predVec, const float* __restrict__ predBias,
    const float* __restrict__ relEmb,
    float* __restrict__ partial, int B)
{
    const int tid  = threadIdx.x;
    const int lane = tid & 31;
    const int w    = tid >> 5;        // wave index in block (0..7)
    const int m    = lane & 15;       // sample row within wave (M dimension)
    const int h    = lane >> 4;       // K-half: lanes 0-15 -> K 0,1 ; 16-31 -> K 2,3

    long s = (long)blockIdx.x * SAMPLES_PER_BLOCK + w * 16 + m;
    const bool in_range = s < (long)B;
    if (!in_range) s = B - 1;         // clamp; excluded from the sum later

    const int li  = lIdx[s];
    const int ri  = rIdx[s];
    const int nli = nlIdx[s];
    const int nri = nrIdx[s];
    const int rel = relIdx[s];

    // float2-unit base pointers for the 5 gathered 128-dim rows.
    const v2f* __restrict__ Lp  = (const v2f*)(predVec + (size_t)li  * DIM);
    const v2f* __restrict__ Rp  = (const v2f*)(predVec + (size_t)ri  * DIM);
    const v2f* __restrict__ NLp = (const v2f*)(predVec + (size_t)nli * DIM);
    const v2f* __restrict__ NRp = (const v2f*)(predVec + (size_t)nri * DIM);
    const v2f* __restrict__ Ep  = (const v2f*)(relEmb  + (size_t)rel * DIM);

    v8f acc0 = {}, acc1 = {}, acc2 = {}, acc3 = {}, acc4 = {}, acc5 = {};
    const v2f ones = {1.0f, 1.0f};

    int off2 = h;                     // float2 index: dims (4j + 2h, 4j + 2h + 1)
    #pragma unroll 4
    for (int j = 0; j < DIM / 4; ++j, off2 += 2) {
        v2f lv = Lp[off2];
        v2f rv = Rp[off2];
        v2f nl = NLp[off2];
        v2f nr = NRp[off2];
        v2f re = Ep[off2];

        v2f dlr = lv - rv;            // ||lv - rv||^2 partial
        v2f dnl = nl - rv;            // ||nlv - rv||^2
        v2f dln = lv - nr;            // ||lv - nrv||^2
        v2f sl  = lv + re;
        v2f tlr = sl - rv;            // ||lv + rel - rv||^2
        v2f tnl = dnl + re;           // ||nlv + rel - rv||^2
        v2f tln = sl - nr;            // ||lv + rel - nrv||^2

        acc0 = wmma_sumsq(dlr * dlr, ones, acc0);
        acc1 = wmma_sumsq(dnl * dnl, ones, acc1);
        acc2 = wmma_sumsq(dln * dln, ones, acc2);
        acc3 = wmma_sumsq(tlr * tlr, ones, acc3);
        acc4 = wmma_sumsq(tnl * tnl, ones, acc4);
        acc5 = wmma_sumsq(tln * tln, ones, acc5);
    }

    // D layout: VGPR r, lanes 0-15 -> row M=r ; lanes 16-31 -> row M=r+8.
    // All 16 columns are equal (B = ones), so lanes 0 and 16 export column 0.
    __shared__ float xfer[BLOCK / 32][6][16];
    if (m == 0) {
        #pragma unroll
        for (int e = 0; e < 8; ++e) {
            xfer[w][0][h * 8 + e] = acc0[e];
            xfer[w][1][h * 8 + e] = acc1[e];
            xfer[w][2][h * 8 + e] = acc2[e];
            xfer[w][3][h * 8 + e] = acc3[e];
            xfer[w][4][h * 8 + e] = acc4[e];
            xfer[w][5][h * 8 + e] = acc5[e];
        }
    }
    __syncthreads();

    float cost = 0.0f;
    if (h == 0 && in_range) {
        const float sd  = xfer[w][0][m];   // ||lv-rv||^2
        const float sdn = xfer[w][1][m];   // ||nlv-rv||^2
        const float sdl = xfer[w][2][m];   // ||lv-nrv||^2
        const float st  = xfer[w][3][m];   // trans lr
        const float stn = xfer[w][4][m];   // trans nl,r
        const float stl = xfer[w][5][m];   // trans l,nr

        const float lb  = predBias[li];
        const float rb  = predBias[ri];
        const float nlb = predBias[nli];
        const float nrb = predBias[nri];

        const bool hypo  = (rel == 4) || (rel == 6);
        const bool hyper = (rel == 3) || (rel == 5);
        const bool syn   = (rel == 0) || (rel == 1) || (rel == 13) || (rel == 17);

        auto score = [&](float d2, float t2, float bd) -> float {
            const float vd = sqrtf(d2);
            float r;
            if (hypo)       r = fmaxf(vd - bd, 0.0f);
            else if (hyper) r = fmaxf(vd + bd, 0.0f);
            else if (syn)   r = vd + fabsf(bd);
            else            r = sqrtf(t2);
            return r;
        };

        const float crt   = score(sd,  st,  lb  - rb);
        const float crtln = score(sdn, stn, nlb - rb);
        const float crtrn = score(sdl, stl, lb  - nrb);

        cost = fmaxf(crt - crtln + MARGIN, 0.0f) +
               fmaxf(crt - crtrn + MARGIN, 0.0f);
    }

    // Deterministic block reduction of per-sample costs.
    __shared__ float sdata[BLOCK];
    sdata[tid] = cost;
    __syncthreads();
    #pragma unroll
    for (int stp = BLOCK / 2; stp > 0; stp >>= 1) {
        if (tid < stp) sdata[tid] += sdata[tid + stp];
        __syncthreads();
    }
    if (tid == 0) partial[blockIdx.x] = sdata[0];
}

__global__ __launch_bounds__(256) void final_reduce_kernel(
    const float* __restrict__ partial, int n, float* __restrict__ out, float invB)
{
    __shared__ float sdata[256];
    float sum = 0.0f;
    for (int i = threadIdx.x; i < n; i += 256) sum += partial[i];
    sdata[threadIdx.x] = sum;
    __syncthreads();
    #pragma unroll
    for (int stp = 128; stp > 0; stp >>= 1) {
        if (threadIdx.x < stp) sdata[threadIdx.x] += sdata[threadIdx.x + stp];
        __syncthreads();
    }
    if (threadIdx.x == 0) out[0] = sdata[0] * invB;
}

extern "C" void kernel_launch(void* const* d_in, const int* in_sizes, int n_in,
                              void* d_out, int out_size, void* d_ws, size_t ws_size,
                              hipStream_t stream) {
    const int*   lIdx    = (const int*)d_in[0];
    const int*   rIdx    = (const int*)d_in[1];
    const int*   relIdx  = (const int*)d_in[2];
    const int*   nlIdx   = (const int*)d_in[3];
    const int*   nrIdx   = (const int*)d_in[4];
    const float* predVec = (const float*)d_in[5];
    const float* predBias= (const float*)d_in[6];
    const float* relEmb  = (const float*)d_in[7];
    float*       out     = (float*)d_out;

    const int B      = in_sizes[0];                         // 262144
    const int blocks = (B + SAMPLES_PER_BLOCK - 1) / SAMPLES_PER_BLOCK;  // 2048
    float* partial   = (float*)d_ws;                        // blocks floats

    crt_cost_kernel<<<blocks, BLOCK, 0, stream>>>(
        lIdx, rIdx, relIdx, nlIdx, nrIdx, predVec, predBias, relEmb, partial, B);
    final_reduce_kernel<<<1, 256, 0, stream>>>(partial, blocks, out, 1.0f / (float)B);
}